// Agg_57380763075323
// MI455X (gfx1250) — compile-verified
//
#include <hip/hip_runtime.h>
#include <stdint.h>

typedef __attribute__((ext_vector_type(2))) float v2f;
typedef __attribute__((ext_vector_type(8))) float v8f;

// Problem shape (fixed by the reference)
constexpr int Bn = 16;
constexpr int Tn = 8192;
constexpr int Dn = 256;
constexpr int Sn = 512;

constexpr int CHUNK  = 16;   // tokens staged per LDS tile
constexpr int STRIDE = Dn;   // unpadded: lets one IOFFSET step both LDS & global sides

// One async b128 issue; OFF (bytes) is added by HW to BOTH the LDS and the
// global address (VGLOBAL async semantics), so consecutive 4-row slices need
// no per-issue address math.
template <int OFF>
__device__ __forceinline__ void async_b128(uint32_t lds_addr, uint64_t gaddr) {
  asm volatile("global_load_async_to_lds_b128 %0, %1, off offset:%2"
               :: "v"(lds_addr), "v"(gaddr), "i"(OFF) : "memory");
}

__device__ __forceinline__ void wait_async_and_barrier() {
  asm volatile("s_wait_asynccnt 0" ::: "memory");
  __syncthreads();
}

__global__ __launch_bounds__(256) void Agg_57380763075323_kernel(
    const float* __restrict__ x,        // [B,T,D]
    const int*   __restrict__ lengths,  // [B]
    const int*   __restrict__ spans,    // [B,S,2]
    float*       __restrict__ out)      // [B,S,3*D]  (min | max | mean)
{
  __shared__ float tile[CHUNK * STRIDE];

  const int bs  = blockIdx.x;          // b*Sn + s
  const int b   = bs / Sn;
  const int s   = bs - b * Sn;
  const int tid = threadIdx.x;         // 0..255, also the feature index f
  const int wave = tid >> 5;           // 0..7
  const int lane = tid & 31;
  const int halfsel = lane >> 4;       // lanes 16..31 carry K=2,3 slots of B
  const int ncol = lane & 15;          // N index within 16x16 tile
  const int fcol0 = wave * 32 + ncol;  // feature column for this wave's chunk 0
  const int fcol1 = fcol0 + 16;        // feature column for this wave's chunk 1

  int start = spans[(b * Sn + s) * 2 + 0];
  int end   = spans[(b * Sn + s) * 2 + 1];
  start = max(start, 0);
  end   = min(end, Tn);
  const int cnt = max(end - start, 0);
  const bool row_ok = (s < lengths[b]) && (cnt > 0);

  const size_t obase = (size_t)bs * (3 * Dn);
  if (!row_ok) {
    // reference leaves these rows zero; harness poisons d_out, so write zeros
    __builtin_nontemporal_store(0.0f, &out[obase + tid]);
    __builtin_nontemporal_store(0.0f, &out[obase + Dn + tid]);
    __builtin_nontemporal_store(0.0f, &out[obase + 2 * Dn + tid]);
    return;
  }

  float vmin =  __builtin_inff();
  float vmax = -__builtin_inff();
  v8f acc0 = {};                 // D accumulator for features [wave*32 .. +15]
  v8f acc1 = {};                 // D accumulator for features [wave*32+16 .. +31]
  const v2f ones = {1.0f, 1.0f}; // A-matrix: all-ones 16x4 (each lane holds 2 K slots)

  // Per-thread copy coordinates: thread covers (row0 + 4k, col4) for k=0..3
  const int row0 = tid >> 6;           // 0..3
  const int col4 = tid & 63;           // float4 column within a 256-float row
  const uint32_t lds0 =
      (uint32_t)(uintptr_t)(&tile[row0 * STRIDE + col4 * 4]);

  const int nchunks = (cnt + CHUNK - 1) / CHUNK;
  for (int ci = 0; ci < nchunks; ++ci) {
    const int tok0 = start + ci * CHUNK;
    const int clen = min(CHUNK, end - tok0);
    const uint64_t g0 = (uint64_t)(uintptr_t)(
        x + ((size_t)(b * Tn + tok0 + row0) * Dn + col4 * 4));

    if (clen == CHUNK) {
      // ---------- fast path: full 16-token chunk, no predication ----------
      async_b128<0>(lds0, g0);                 // rows 0..3
      async_b128<4 * Dn * 4>(lds0, g0);        // rows 4..7   (+4096B both sides)
      async_b128<8 * Dn * 4>(lds0, g0);        // rows 8..11
      async_b128<12 * Dn * 4>(lds0, g0);       // rows 12..15
      wait_async_and_barrier();

      // min / max: one feature per thread
#pragma unroll
      for (int t = 0; t < CHUNK; ++t) {
        const float v = tile[t * STRIDE + tid];
        vmin = fminf(vmin, v);
        vmax = fmaxf(vmax, v);
      }

      // token sum via WMMA: D += ones(16x4) x B(4x16), 4 K-chunks of 4 tokens
#pragma unroll
      for (int tc = 0; tc < 4; ++tc) {
        const int ta = tc * 4 + halfsel * 2;
        const int tb = ta + 1;
        v2f b0, b1;
        b0.x = tile[ta * STRIDE + fcol0];
        b0.y = tile[tb * STRIDE + fcol0];
        b1.x = tile[ta * STRIDE + fcol1];
        b1.y = tile[tb * STRIDE + fcol1];
        acc0 = __builtin_amdgcn_wmma_f32_16x16x4_f32(
            false, ones, false, b0, (short)0, acc0, false, false);
        acc1 = __builtin_amdgcn_wmma_f32_16x16x4_f32(
            false, ones, false, b1, (short)0, acc1, false, false);
      }
    } else {
      // ---------- tail path: partial chunk, predicated per 4-row slice ----------
      if (row0 + 0 < clen) async_b128<0>(lds0, g0);
      if (row0 + 4 < clen) async_b128<4 * Dn * 4>(lds0, g0);
      if (row0 + 8 < clen) async_b128<8 * Dn * 4>(lds0, g0);
      if (row0 + 12 < clen) async_b128<12 * Dn * 4>(lds0, g0);
      wait_async_and_barrier();

      for (int t = 0; t < clen; ++t) {
        const float v = tile[t * STRIDE + tid];
        vmin = fminf(vmin, v);
        vmax = fmaxf(vmax, v);
      }

#pragma unroll
      for (int tc = 0; tc < 4; ++tc) {
        const int ta = tc * 4 + halfsel * 2;
        const int tb = ta + 1;
        v2f b0, b1;
        b0.x = (ta < clen) ? tile[ta * STRIDE + fcol0] : 0.0f;
        b0.y = (tb < clen) ? tile[tb * STRIDE + fcol0] : 0.0f;
        b1.x = (ta < clen) ? tile[ta * STRIDE + fcol1] : 0.0f;
        b1.y = (tb < clen) ? tile[tb * STRIDE + fcol1] : 0.0f;
        acc0 = __builtin_amdgcn_wmma_f32_16x16x4_f32(
            false, ones, false, b0, (short)0, acc0, false, false);
        acc1 = __builtin_amdgcn_wmma_f32_16x16x4_f32(
            false, ones, false, b1, (short)0, acc1, false, false);
      }
    }
    __syncthreads();  // tile reused next chunk
  }

  // ---------- writeback (non-temporal: 24MB write-once stream) ----------
  __builtin_nontemporal_store(vmin, &out[obase + tid]);
  __builtin_nontemporal_store(vmax, &out[obase + Dn + tid]);
  const float inv = 1.0f / (float)cnt;
  if (lane < 16) {
    // acc[0] on lanes 0..15 = D[M=0, N=lane] = token-sum for that feature
    __builtin_nontemporal_store(acc0[0] * inv, &out[obase + 2 * Dn + fcol0]);
    __builtin_nontemporal_store(acc1[0] * inv, &out[obase + 2 * Dn + fcol1]);
  }
}

extern "C" void kernel_launch(void* const* d_in, const int* in_sizes, int n_in,
                              void* d_out, int out_size, void* d_ws, size_t ws_size,
                              hipStream_t stream) {
  const float* x       = (const float*)d_in[0];
  const int*   lengths = (const int*)d_in[1];
  const int*   spans   = (const int*)d_in[2];
  float*       out     = (float*)d_out;

  dim3 grid(Bn * Sn);   // one block per (b, s) span
  dim3 block(256);      // 8 wave32 waves
  Agg_57380763075323_kernel<<<grid, block, 0, stream>>>(x, lengths, spans, out);
}